// Sage_8967891714111
// MI455X (gfx1250) — compile-verified
//
#include <hip/hip_runtime.h>
#include <math.h>

typedef float v2f __attribute__((ext_vector_type(2)));
typedef float v8f __attribute__((ext_vector_type(8)));

#define NNODES 100000
#define NEDGES 600000
#define NFEAT  128
#define NHID   64
#define NCLS   40

// ---------------------------------------------------------------- utilities
// float4 zeroing (count must be a multiple of 4 — all our buffers are)
__global__ void zero_f32x4_kernel(float4* __restrict__ p, int n4) {
  int stride = gridDim.x * blockDim.x;
  float4 z = {0.0f, 0.0f, 0.0f, 0.0f};
  for (int i = blockIdx.x * blockDim.x + threadIdx.x; i < n4; i += stride)
    p[i] = z;
}

__global__ void degree_kernel(const int* __restrict__ ei, float* __restrict__ deg) {
  int e = blockIdx.x * blockDim.x + threadIdx.x;
  if (e < NEDGES) atomicAdd(&deg[ei[NEDGES + e]], 1.0f);
}

// scatter-sum: msg[dst] += feat[src], FEAT/4 threads per edge, float4 gathers
template <int FEAT>
__global__ void scatter_sum_kernel(const float* __restrict__ feat,
                                   const int* __restrict__ ei,
                                   float* __restrict__ msg) {
  const int per_edge = FEAT / 4;
  int gid = blockIdx.x * blockDim.x + threadIdx.x;
  int e = gid / per_edge;
  if (e >= NEDGES) return;
  int c = (gid % per_edge) * 4;
  int s = ei[e];
  int d = ei[NEDGES + e];
  float4 v = *(const float4*)(feat + (size_t)s * FEAT + c);
  float* o = msg + (size_t)d * FEAT + c;
  atomicAdd(o + 0, v.x);
  atomicAdd(o + 1, v.y);
  atomicAdd(o + 2, v.z);
  atomicAdd(o + 3, v.w);
}

// ------------------------------------------------------ layer 1 fused GEMM
// out16x64 = (msg1/deg) @ W1l + b1 + x @ W1r, sigmoid -> h
// block = 128 threads (4 waves), one 16-node tile per block, wave w owns
// columns [16w, 16w+16). fp32 WMMA 16x16x4, K-loop of 32 steps x 2 matmuls.
#define STR1 132  // 132 % 64 == 4 -> conflict-free column access
__global__ __launch_bounds__(128) void sage_layer1_kernel(
    const float* __restrict__ msg1, const float* __restrict__ deg,
    const float* __restrict__ x,
    const float* __restrict__ Wl, const float* __restrict__ bvec,
    const float* __restrict__ Wr, float* __restrict__ hout) {
  __shared__ float As[16 * STR1];  // scaled aggregate
  __shared__ float Xs[16 * STR1];  // self features
  const size_t base = (size_t)blockIdx.x * 16;
  const int t = threadIdx.x;
  const int lrow = t >> 3;
  const int lq = t & 7;
  const float inv = 1.0f / fmaxf(deg[base + lrow], 1.0f);
  const float* mrow = msg1 + (base + lrow) * NFEAT;
  const float* xrow = x + (base + lrow) * NFEAT;
#pragma unroll
  for (int cb = 0; cb < 4; ++cb) {
    int col = cb * 32 + lq * 4;
    float4 mv = *(const float4*)(mrow + col);
    float4 xv = *(const float4*)(xrow + col);
    float* ap = &As[lrow * STR1 + col];
    ap[0] = mv.x * inv; ap[1] = mv.y * inv; ap[2] = mv.z * inv; ap[3] = mv.w * inv;
    float* xp = &Xs[lrow * STR1 + col];
    xp[0] = xv.x; xp[1] = xv.y; xp[2] = xv.z; xp[3] = xv.w;
  }
  __syncthreads();

  const int lane = t & 31;
  const int wv = t >> 5;                 // n-tile 0..3
  const int m = lane & 15;               // A row
  const int koff = (lane >> 4) * 2;      // K sub-offset for upper half-wave
  const int n = wv * 16 + (lane & 15);   // B/C column
  v8f acc = {};
#pragma unroll 4
  for (int kk = 0; kk < NFEAT / 4; ++kk) {
    const int k0 = kk * 4 + koff;
    v2f a, b;
    // agg @ W1l
    a.x = As[m * STR1 + k0];
    a.y = As[m * STR1 + k0 + 1];
    b.x = Wl[k0 * NHID + n];
    b.y = Wl[(k0 + 1) * NHID + n];
    acc = __builtin_amdgcn_wmma_f32_16x16x4_f32(false, a, false, b, (short)0, acc, false, false);
    // x @ W1r
    a.x = Xs[m * STR1 + k0];
    a.y = Xs[m * STR1 + k0 + 1];
    b.x = Wr[k0 * NHID + n];
    b.y = Wr[(k0 + 1) * NHID + n];
    acc = __builtin_amdgcn_wmma_f32_16x16x4_f32(false, a, false, b, (short)0, acc, false, false);
  }
  const float bb = bvec[n];
  const int mb = (lane >> 4) * 8;  // C/D: VGPR r holds M=r (lanes<16) / M=r+8
#pragma unroll
  for (int r = 0; r < 8; ++r) {
    float v = acc[r] + bb;
    // sigmoid via v_exp + v_rcp (avoids IEEE div_scale/div_fmas sequence)
    v = __builtin_amdgcn_rcpf(1.0f + __expf(-v));
    hout[(base + mb + r) * NHID + n] = v;
  }
}

// ------------------------------------------------------ layer 2 fused GEMM
// out16x40 = (msg2/deg) @ W2l + b2 + h @ W2r
// 4 waves; wave 3 (cols 48..63) computes a discarded all-zero tile so EXEC
// stays all-ones across every WMMA. B loads / stores masked to n < 40.
#define STR2 68  // 68 % 64 == 4 -> conflict-free
__global__ __launch_bounds__(128) void sage_layer2_kernel(
    const float* __restrict__ msg2, const float* __restrict__ deg,
    const float* __restrict__ hbuf,
    const float* __restrict__ Wl, const float* __restrict__ bvec,
    const float* __restrict__ Wr, float* __restrict__ out) {
  __shared__ float As[16 * STR2];
  __shared__ float Hs[16 * STR2];
  const size_t base = (size_t)blockIdx.x * 16;
  const int t = threadIdx.x;
  const int lrow = t >> 3;
  const int lq = t & 7;
  const float inv = 1.0f / fmaxf(deg[base + lrow], 1.0f);
  const float* mrow = msg2 + (base + lrow) * NHID;
  const float* hrow = hbuf + (base + lrow) * NHID;
#pragma unroll
  for (int cb = 0; cb < 2; ++cb) {
    int col = cb * 32 + lq * 4;
    float4 mv = *(const float4*)(mrow + col);
    float4 hv = *(const float4*)(hrow + col);
    float* ap = &As[lrow * STR2 + col];
    ap[0] = mv.x * inv; ap[1] = mv.y * inv; ap[2] = mv.z * inv; ap[3] = mv.w * inv;
    float* hp = &Hs[lrow * STR2 + col];
    hp[0] = hv.x; hp[1] = hv.y; hp[2] = hv.z; hp[3] = hv.w;
  }
  __syncthreads();

  const int lane = t & 31;
  const int wv = t >> 5;
  const int m = lane & 15;
  const int koff = (lane >> 4) * 2;
  const int n = wv * 16 + (lane & 15);
  const bool nv = (n < NCLS);
  v8f acc = {};
#pragma unroll 4
  for (int kk = 0; kk < NHID / 4; ++kk) {
    const int k0 = kk * 4 + koff;
    v2f a, b;
    a.x = As[m * STR2 + k0];
    a.y = As[m * STR2 + k0 + 1];
    b.x = nv ? Wl[k0 * NCLS + n] : 0.0f;
    b.y = nv ? Wl[(k0 + 1) * NCLS + n] : 0.0f;
    acc = __builtin_amdgcn_wmma_f32_16x16x4_f32(false, a, false, b, (short)0, acc, false, false);
    a.x = Hs[m * STR2 + k0];
    a.y = Hs[m * STR2 + k0 + 1];
    b.x = nv ? Wr[k0 * NCLS + n] : 0.0f;
    b.y = nv ? Wr[(k0 + 1) * NCLS + n] : 0.0f;
    acc = __builtin_amdgcn_wmma_f32_16x16x4_f32(false, a, false, b, (short)0, acc, false, false);
  }
  if (nv) {
    const float bb = bvec[n];
    const int mb = (lane >> 4) * 8;
#pragma unroll
    for (int r = 0; r < 8; ++r)
      out[(base + mb + r) * NCLS + n] = acc[r] + bb;
  }
}

// ---------------------------------------------------------------- launcher
extern "C" void kernel_launch(void* const* d_in, const int* in_sizes, int n_in,
                              void* d_out, int out_size, void* d_ws, size_t ws_size,
                              hipStream_t stream) {
  (void)in_sizes; (void)n_in; (void)out_size; (void)ws_size;
  const float* x   = (const float*)d_in[0];
  const int*   ei  = (const int*)d_in[1];  // [2, E]: row0=src, row1=dst
  const float* W1l = (const float*)d_in[2];
  const float* b1  = (const float*)d_in[3];
  const float* W1r = (const float*)d_in[4];
  const float* W2l = (const float*)d_in[5];
  const float* b2  = (const float*)d_in[6];
  const float* W2r = (const float*)d_in[7];
  float* out = (float*)d_out;

  // workspace carve-out (~103 MB)
  char* ws = (char*)d_ws;
  size_t off = 0;
  float* msg1 = (float*)(ws + off); off += (size_t)NNODES * NFEAT * sizeof(float);
  float* deg  = (float*)(ws + off); off += (((size_t)NNODES * sizeof(float)) + 255) & ~(size_t)255;
  float* hbuf = (float*)(ws + off); off += (size_t)NNODES * NHID * sizeof(float);
  float* msg2 = (float*)(ws + off); off += (size_t)NNODES * NHID * sizeof(float);

  const int ZB = 256, ZG = 2048;
  // ---- layer 1
  zero_f32x4_kernel<<<ZG, ZB, 0, stream>>>((float4*)msg1, NNODES * NFEAT / 4);
  zero_f32x4_kernel<<<ZG, ZB, 0, stream>>>((float4*)deg, NNODES / 4);
  degree_kernel<<<(NEDGES + 255) / 256, 256, 0, stream>>>(ei, deg);
  scatter_sum_kernel<NFEAT>
      <<<(NEDGES * (NFEAT / 4) + 255) / 256, 256, 0, stream>>>(x, ei, msg1);
  sage_layer1_kernel<<<NNODES / 16, 128, 0, stream>>>(msg1, deg, x, W1l, b1, W1r, hbuf);
  // ---- layer 2
  zero_f32x4_kernel<<<ZG, ZB, 0, stream>>>((float4*)msg2, NNODES * NHID / 4);
  scatter_sum_kernel<NHID>
      <<<(NEDGES * (NHID / 4) + 255) / 256, 256, 0, stream>>>(hbuf, ei, msg2);
  sage_layer2_kernel<<<NNODES / 16, 128, 0, stream>>>(msg2, deg, hbuf, W2l, b2, W2r, out);
}